// GCN_Block_65747359367659
// MI455X (gfx1250) — compile-verified
//
#include <hip/hip_runtime.h>

typedef float v2f __attribute__((ext_vector_type(2)));
typedef float v8f __attribute__((ext_vector_type(8)));

#define N_  32
#define C_  256
#define T_  64
#define V_  25
#define K_  3
#define D_  256
#define KT_ 9
#define NT_ (N_*T_)          // 2048 (n,t) pairs
#define M_  (NT_*V_)         // 51200 flattened (n,t,v)
#define TV_ (T_*V_)          // 1600 per-batch columns
#define EPS_ 1e-5f
#define PITCH_ 132           // float4-row pitch for K-grouped LDS tiles

static __device__ __forceinline__ v8f wmma_f32(v2f a, v2f b, v8f c) {
  // D = A(16x4 f32) * B(4x16 f32) + C(16x16 f32)
  return __builtin_amdgcn_wmma_f32_16x16x4_f32(false, a, false, b, (short)0, c,
                                               false, false);
}

// LDS tile layout: laf[((kq*PITCH_ + row) << 2) + comp] = A(row, 4*kq + comp)
// WMMA fragment (kk = ks + 2*hi, hi = lane>>4) = 8 contiguous bytes:
//   comp (2*hi, 2*hi+1) of float4 [kq][row]  ->  single ds_load_b64, no movs.
static __device__ __forceinline__ v2f frag_ld(const float* lf, int kq, int row,
                                              int hi) {
  return *(const v2f*)&lf[(((kq * PITCH_) + row) << 2) + (hi << 1)];
}

// ---------------------------------------------------------------------------
// k0: bias_eff[d][w] = sum_k b_gcn[k,d] * sum_v A[k,v,w]; zero BN accumulators
// ---------------------------------------------------------------------------
__global__ void k0_bias_stats(const float* __restrict__ A,
                              const float* __restrict__ b_gcn,
                              float* __restrict__ bias_eff,
                              float* __restrict__ gsum,
                              float* __restrict__ gsq) {
  int d = threadIdx.x;
  gsum[d] = 0.f;
  gsq[d]  = 0.f;
  for (int w = 0; w < V_; ++w) {
    float acc = 0.f;
    for (int k = 0; k < K_; ++k) {
      float cs = 0.f;
      for (int v = 0; v < V_; ++v) cs += A[(k * V_ + v) * V_ + w];
      acc += b_gcn[k * D_ + d] * cs;
    }
    bias_eff[d * V_ + w] = acc;
  }
}

// ---------------------------------------------------------------------------
// k1: Xh_k[c][nt*25+w] = sum_v x[n,c,t,v] * A[k,v,w]   (graph aggregation)
// ---------------------------------------------------------------------------
__global__ void __launch_bounds__(256)
k1_xA(const float* __restrict__ x, const float* __restrict__ A, int k,
      float* __restrict__ xh) {
  __shared__ float sA[V_ * V_];
  for (int i = threadIdx.x; i < V_ * V_; i += 256) sA[i] = A[k * V_ * V_ + i];
  __syncthreads();

  int g  = blockIdx.x * 256 + threadIdx.x;  // g = c*NT + nt
  int c  = g >> 11;                         // /2048
  int nt = g & 2047;
  int n  = nt >> 6, t = nt & 63;
  const float* xp = x + ((size_t)(n * C_ + c) * T_ + t) * V_;
  float xv[V_];
#pragma unroll
  for (int v = 0; v < V_; ++v) xv[v] = xp[v];
  float* op = xh + (size_t)c * M_ + nt * V_;
#pragma unroll
  for (int w = 0; w < V_; ++w) {
    float acc = 0.f;
#pragma unroll
    for (int v = 0; v < V_; ++v) acc = fmaf(xv[v], sA[v * V_ + w], acc);
    op[w] = acc;
  }
}

// ---------------------------------------------------------------------------
// k2: y2[d][m] (k==0 ? = bias_eff : +=) W_gcn[k] (256x256) @ Xh_k (256x51200)
//     128x128 WG tile, 8 waves of 64x32, fp32 WMMA 16x16x4.
//     Register-pipelined: global loads for step i+1 issue before compute of
//     step i; s_wait_loadcnt is covered by the 32-WMMA block.
// ---------------------------------------------------------------------------
__global__ void __launch_bounds__(256)
k2_gcn_gemm(const float* __restrict__ Wg_all,    // [K][D][C]
            const float* __restrict__ xh,        // [C][M]
            const float* __restrict__ bias_eff,  // [D][V]
            float* __restrict__ y2,              // [D][M]
            int k) {
  __shared__ float laf[4 * PITCH_ * 4];  // A: 4 k-quads x 128 rows (float4)
  __shared__ float lbf[4 * PITCH_ * 4];  // B: 4 k-quads x 128 cols (float4)
  const float* Wg = Wg_all + (size_t)k * D_ * C_;
  int tid  = threadIdx.x;
  int lane = tid & 31, wave = tid >> 5;
  int hi = lane >> 4, ln = lane & 15;
  int wr = wave >> 2, wc = wave & 3;     // wave grid 2x4
  int d0 = blockIdx.y * 128;
  int m0 = blockIdx.x * 128;
  v8f acc[4][2] = {};

  float4 ra[2];
  float  rb[8];
  auto load_regs = [&](int c0) {
#pragma unroll
    for (int i = 0; i < 2; ++i) {        // A tile: 2x global_load_b128
      int idx = tid + i * 256;           // 512 float4 chunks
      ra[i] = *(const float4*)&Wg[(d0 + (idx >> 2)) * C_ + c0 + ((idx & 3) << 2)];
    }
#pragma unroll
    for (int i = 0; i < 8; ++i) {        // B tile: 16 x 128 cols
      int idx = tid + i * 256;
      rb[i] = xh[(size_t)(c0 + (idx >> 7)) * M_ + m0 + (idx & 127)];
    }
    if (c0 + 16 < C_) {                  // prefetch one step further (gfx1250)
      __builtin_prefetch(
          &xh[(size_t)(c0 + 16 + (tid >> 4)) * M_ + m0 + ((tid & 15) << 3)], 0,
          0);
    }
  };
  auto store_lds = [&]() {
#pragma unroll
    for (int i = 0; i < 2; ++i) {        // ds_store_b128
      int idx = tid + i * 256;
      ((float4*)laf)[(idx & 3) * PITCH_ + (idx >> 2)] = ra[i];
    }
#pragma unroll
    for (int i = 0; i < 8; ++i) {
      int idx = tid + i * 256;
      int col = idx & 127, rr = idx >> 7;
      lbf[(((rr >> 2) * PITCH_ + col) << 2) + (rr & 3)] = rb[i];
    }
  };

  load_regs(0);
  store_lds();
  __syncthreads();
  for (int c0 = 0; c0 < C_; c0 += 16) {
    bool more = (c0 + 16 < C_);
    if (more) load_regs(c0 + 16);        // global loads in flight during WMMAs
#pragma unroll
    for (int ks = 0; ks < 16; ks += 4) {
      int kq = ks >> 2;
      v2f a[4], b[2];
#pragma unroll
      for (int i = 0; i < 4; ++i)
        a[i] = frag_ld(laf, kq, wr * 64 + i * 16 + ln, hi);
#pragma unroll
      for (int j = 0; j < 2; ++j)
        b[j] = frag_ld(lbf, kq, wc * 32 + j * 16 + ln, hi);
#pragma unroll
      for (int i = 0; i < 4; ++i)
#pragma unroll
        for (int j = 0; j < 2; ++j)
          acc[i][j] = wmma_f32(a[i], b[j], acc[i][j]);
    }
    if (more) {
      __syncthreads();                   // all waves done reading the buffer
      store_lds();
      __syncthreads();                   // new tile visible
    }
  }

#pragma unroll
  for (int i = 0; i < 4; ++i) {
#pragma unroll
    for (int j = 0; j < 2; ++j) {
      int m = m0 + wc * 32 + j * 16 + ln;
#pragma unroll
      for (int p = 0; p < 8; ++p) {
        int d = d0 + wr * 64 + i * 16 + p + 8 * hi;
        size_t o = (size_t)d * M_ + m;
        float base = (k == 0) ? bias_eff[d * V_ + (m % V_)] : y2[o];
        y2[o] = base + acc[i][j][p];
      }
    }
  }
}

// ---------------------------------------------------------------------------
// k3: TCN  z[n,d,t,v] = b_tcn[d] + sum_{tau,c} Wt[d,c,tau]*y2[c][n,t+tau-4,v]
//     GEMM per n, reduction r = tau*256 + c (2304), zero-padded temporal shift
//     applied in the B-tile load. Register-pipelined like k2. Also accumulates
//     BN sum / sumsq per channel.
// ---------------------------------------------------------------------------
__global__ void __launch_bounds__(256)
k3_tcn_gemm(const float* __restrict__ Wt,     // [D][D][KT]
            const float* __restrict__ b_tcn,  // [D]
            const float* __restrict__ y2,     // [C][N*T*V]
            float* __restrict__ zb,           // [N][D][T*V]
            float* __restrict__ gsum, float* __restrict__ gsq) {
  __shared__ float laf[4 * PITCH_ * 4];
  __shared__ float lbf[4 * PITCH_ * 4];
  __shared__ float s_sum[128];
  __shared__ float s_sq[128];
  int tid = threadIdx.x;
  if (tid < 128) { s_sum[tid] = 0.f; s_sq[tid] = 0.f; }
  int lane = tid & 31, wave = tid >> 5;
  int hi = lane >> 4, ln = lane & 15;
  int wr = wave >> 2, wc = wave & 3;
  int d0  = blockIdx.y * 128;
  int tv0 = blockIdx.x * 128;   // columns (t*25+v), padded past 1600
  int n   = blockIdx.z;
  v8f acc[4][2] = {};

  float rA[8], rB[8];
  auto load_regs = [&](int r0) {
    int tau = r0 >> 8;          // chunks of 16 never cross tau (256 % 16 == 0)
    int c0  = r0 & 255;
    int shift = (tau - 4) * V_; // PAD = 4
#pragma unroll
    for (int i = 0; i < 8; ++i) {  // A tile: Wt[d][c][tau] (stride-9 gather)
      int idx = tid + i * 256;
      rA[i] = Wt[(size_t)((d0 + (idx >> 4)) * D_ + c0 + (idx & 15)) * KT_ + tau];
    }
#pragma unroll
    for (int i = 0; i < 8; ++i) {  // B tile: shifted, zero-padded y2
      int idx = tid + i * 256;
      int col = idx & 127, rr = idx >> 7;
      int tv = tv0 + col;
      int stv = tv + shift;
      float v = 0.f;
      if (tv < TV_ && stv >= 0 && stv < TV_)
        v = y2[(size_t)(c0 + rr) * M_ + n * TV_ + stv];
      rB[i] = v;
    }
  };
  auto store_lds = [&]() {
#pragma unroll
    for (int i = 0; i < 8; ++i) {
      int idx = tid + i * 256;
      int cc = idx & 15, row = idx >> 4;
      laf[(((cc >> 2) * PITCH_ + row) << 2) + (cc & 3)] = rA[i];
    }
#pragma unroll
    for (int i = 0; i < 8; ++i) {
      int idx = tid + i * 256;
      int col = idx & 127, rr = idx >> 7;
      lbf[(((rr >> 2) * PITCH_ + col) << 2) + (rr & 3)] = rB[i];
    }
  };

  load_regs(0);
  store_lds();
  __syncthreads();
  for (int r0 = 0; r0 < D_ * KT_; r0 += 16) {
    bool more = (r0 + 16 < D_ * KT_);
    if (more) load_regs(r0 + 16);        // overlap with WMMA block
#pragma unroll
    for (int ks = 0; ks < 16; ks += 4) {
      int kq = ks >> 2;
      v2f a[4], b[2];
#pragma unroll
      for (int i = 0; i < 4; ++i)
        a[i] = frag_ld(laf, kq, wr * 64 + i * 16 + ln, hi);
#pragma unroll
      for (int j = 0; j < 2; ++j)
        b[j] = frag_ld(lbf, kq, wc * 32 + j * 16 + ln, hi);
#pragma unroll
      for (int i = 0; i < 4; ++i)
#pragma unroll
        for (int j = 0; j < 2; ++j)
          acc[i][j] = wmma_f32(a[i], b[j], acc[i][j]);
    }
    if (more) {
      __syncthreads();
      store_lds();
      __syncthreads();
    }
  }

  // epilogue: +bias, store z, per-channel sum/sumsq (shfl reduce -> LDS -> global)
#pragma unroll
  for (int i = 0; i < 4; ++i) {
#pragma unroll
    for (int p = 0; p < 8; ++p) {
      int row = wr * 64 + i * 16 + p + 8 * hi;  // local channel row [0,128)
      int d = d0 + row;
      float bt = b_tcn[d];
      float s = 0.f, q = 0.f;
#pragma unroll
      for (int j = 0; j < 2; ++j) {
        int tv = tv0 + wc * 32 + j * 16 + ln;
        if (tv < TV_) {
          float val = acc[i][j][p] + bt;
          zb[(size_t)(n * D_ + d) * TV_ + tv] = val;
          s += val;
          q += val * val;
        }
      }
#pragma unroll
      for (int off = 1; off < 16; off <<= 1) {  // reduce within 16-lane half
        s += __shfl_xor(s, off);
        q += __shfl_xor(q, off);
      }
      if (ln == 0) {
        atomicAdd(&s_sum[row], s);
        atomicAdd(&s_sq[row], q);
      }
    }
  }
  __syncthreads();
  if (tid < 128) {
    atomicAdd(&gsum[d0 + tid], s_sum[tid]);
    atomicAdd(&gsq[d0 + tid], s_sq[tid]);
  }
}

// ---------------------------------------------------------------------------
// k4: BatchNorm (batch stats) + residual + ReLU
// ---------------------------------------------------------------------------
__global__ void __launch_bounds__(256)
k4_bn_relu(const float* __restrict__ zb, const float* __restrict__ x,
           const float* __restrict__ gsum, const float* __restrict__ gsq,
           const float* __restrict__ gamma, const float* __restrict__ beta,
           float* __restrict__ out) {
  size_t idx = (size_t)blockIdx.x * 256 + threadIdx.x;  // [n][d][tv]
  int d = (int)((idx / TV_) & (D_ - 1));
  const float inv_cnt = 1.0f / (float)M_;
  float mu  = gsum[d] * inv_cnt;
  float var = gsq[d] * inv_cnt - mu * mu;
  float z = zb[idx];
  float r = gamma[d] * (z - mu) * rsqrtf(var + EPS_) + beta[d] + x[idx];
  out[idx] = fmaxf(r, 0.f);
}

// ---------------------------------------------------------------------------
extern "C" void kernel_launch(void* const* d_in, const int* in_sizes, int n_in,
                              void* d_out, int out_size, void* d_ws,
                              size_t ws_size, hipStream_t stream) {
  const float* x     = (const float*)d_in[0];
  const float* A     = (const float*)d_in[1];
  const float* Wg    = (const float*)d_in[2];
  const float* bg    = (const float*)d_in[3];
  const float* Wt    = (const float*)d_in[4];
  const float* bt    = (const float*)d_in[5];
  const float* gamma = (const float*)d_in[6];
  const float* beta  = (const float*)d_in[7];
  float* out = (float*)d_out;

  float* ws = (float*)d_ws;                 // workspace layout (floats):
  float* xh = ws;                           // [C][M]      13,107,200
  float* y2 = xh + (size_t)C_ * M_;         // [D][M]      13,107,200
  float* zb = y2 + (size_t)D_ * M_;         // [N][D][TV]  13,107,200
  float* be = zb + (size_t)D_ * M_;         // [D][V]      6,400
  float* gs = be + D_ * V_;                 // [D]         256
  float* gq = gs + D_;                      // [D]         256

  k0_bias_stats<<<1, 256, 0, stream>>>(A, bg, be, gs, gq);
  for (int k = 0; k < K_; ++k) {
    k1_xA<<<(NT_ * C_) / 256, 256, 0, stream>>>(x, A, k, xh);
    k2_gcn_gemm<<<dim3(M_ / 128, D_ / 128), 256, 0, stream>>>(Wg, xh, be, y2, k);
  }
  k3_tcn_gemm<<<dim3((TV_ + 127) / 128, D_ / 128, N_), 256, 0, stream>>>(
      Wt, bt, y2, zb, gs, gq);
  k4_bn_relu<<<(size_t)(N_ * D_ * TV_) / 256, 256, 0, stream>>>(
      zb, x, gs, gq, gamma, beta, out);
}